// DGLBatchCapsuleLayer_28260884808233
// MI455X (gfx1250) — compile-verified
//
#include <hip/hip_runtime.h>
#include <math.h>

#define IN_NUM  1152
#define IN_DIM  8
#define OUT_NUM 10
#define OUT_DIM 16
#define BATCH   256
#define XK      (IN_DIM * IN_NUM)        // 9216
#define PN      (OUT_NUM * OUT_DIM)      // 160
#define NSPLIT  9                        // i-splits of 128 for s-GEMM
#define S_ELEMS (OUT_NUM * BATCH * OUT_DIM)  // 40960

typedef __attribute__((ext_vector_type(2))) float v2f;
typedef __attribute__((ext_vector_type(8))) float v8f;

__device__ __forceinline__ v8f wmma_f32_k4(v2f a, v2f b, v8f c) {
  // D(16x16,f32) = A(16x4,f32) x B(4x16,f32) + C
  return __builtin_amdgcn_wmma_f32_16x16x4_f32(false, a, false, b, (short)0, c,
                                               false, false);
}

__global__ void k_zero(float* __restrict__ p, int n) {
  int t = blockIdx.x * blockDim.x + threadIdx.x;
  if (t < n) p[t] = 0.0f;
}

// c[i,j] = softmax over j of b[i,j]
__global__ void k_softmax(const float* __restrict__ bij, float* __restrict__ c) {
  int i = blockIdx.x * blockDim.x + threadIdx.x;
  if (i >= IN_NUM) return;
  float mx = -3.402823e38f;
  for (int j = 0; j < OUT_NUM; ++j) mx = fmaxf(mx, bij[i * OUT_NUM + j]);
  float e[OUT_NUM];
  float sum = 0.0f;
  for (int j = 0; j < OUT_NUM; ++j) {
    e[j] = __expf(bij[i * OUT_NUM + j] - mx);
    sum += e[j];
  }
  float inv = 1.0f / sum;
  for (int j = 0; j < OUT_NUM; ++j) c[i * OUT_NUM + j] = e[j] * inv;
}

// Partial s: sp[z][j,b,o] = sum_{i in split z, d} c[i,j] * W[i,j,o,d] * x[b,d,i]
// One wave = one 16(batch) x 16(out_dim) tile for capsule j; deterministic
// (no atomics) — each i-split writes a private slab, reduced afterwards.
__global__ void __launch_bounds__(128)
k_s_gemm(const float* __restrict__ x, const float* __restrict__ W,
         const float* __restrict__ c, float* __restrict__ sp) {
  const int lane = threadIdx.x & 31;
  const int wave = threadIdx.x >> 5;
  const int mn = lane & 15;   // A: batch row m; B: out_dim col n
  const int hi = lane >> 4;   // K-pair select
  const int bt = (blockIdx.x * 4 + wave) * 16;  // batch tile base
  const int j  = blockIdx.y;
  const int i0 = blockIdx.z * 128;

  v8f acc = {};
  const float* xrow = x + (bt + mn) * XK;  // + k*IN_NUM + i
  const int kb  = 2 * hi;       // k = {0,1} or {2,3}
  const int kb2 = 4 + 2 * hi;   // k = {4,5} or {6,7}
  for (int i = i0; i < i0 + 128; ++i) {
    const float cw = c[i * OUT_NUM + j];
    const float* xp = xrow + i;
    const float* wp = W + ((i * OUT_NUM + j) * OUT_DIM + mn) * IN_DIM;
    v2f a0, b0, a1, b1;
    a0.x = xp[kb * IN_NUM];
    a0.y = xp[(kb + 1) * IN_NUM];
    b0.x = cw * wp[kb];
    b0.y = cw * wp[kb + 1];
    acc = wmma_f32_k4(a0, b0, acc);
    a1.x = xp[kb2 * IN_NUM];
    a1.y = xp[(kb2 + 1) * IN_NUM];
    b1.x = cw * wp[kb2];
    b1.y = cw * wp[kb2 + 1];
    acc = wmma_f32_k4(a1, b1, acc);
  }
  float* out = sp + (size_t)blockIdx.z * S_ELEMS;
  for (int r = 0; r < 8; ++r) {
    int row = bt + r + 8 * hi;
    out[(j * BATCH + row) * OUT_DIM + mn] = acc[r];
  }
}

// s[t] = sum_z sp[z][t]  (fixed-order, deterministic)
__global__ void k_reduce_s(const float* __restrict__ sp, float* __restrict__ s) {
  int t = blockIdx.x * blockDim.x + threadIdx.x;
  if (t >= S_ELEMS) return;
  float sum = 0.0f;
  for (int z = 0; z < NSPLIT; ++z) sum += sp[(size_t)z * S_ELEMS + t];
  s[t] = sum;
}

// v = squash(s) along out_dim
__global__ void k_squash(const float* __restrict__ s, float* __restrict__ v) {
  int idx = blockIdx.x * blockDim.x + threadIdx.x;  // over OUT_NUM*BATCH
  if (idx >= OUT_NUM * BATCH) return;
  const float* spt = s + idx * OUT_DIM;
  float* vp = v + idx * OUT_DIM;
  float magsq = 0.0f;
  for (int o = 0; o < OUT_DIM; ++o) magsq += spt[o] * spt[o];
  float mag = sqrtf(magsq);
  float scale = magsq / ((1.0f + magsq) * mag);
  for (int o = 0; o < OUT_DIM; ++o) vp[o] = spt[o] * scale;
}

// P[(i*8+d), (j*16+o)] = sum_b x[b,d,i] * v[j,b,o]   (M=9216, N=160, K=256)
__global__ void __launch_bounds__(128)
k_p_gemm(const float* __restrict__ x, const float* __restrict__ v,
         float* __restrict__ P) {
  const int lane = threadIdx.x & 31;
  const int wave = threadIdx.x >> 5;
  const int mn = lane & 15;
  const int hi = lane >> 4;
  const int r0 = (blockIdx.x * 4 + wave) * 16;  // row tile base in (i,d)
  const int j  = blockIdx.y;
  const int i  = (r0 + mn) >> 3;
  const int d  = (r0 + mn) & 7;
  const float* xp = x + d * IN_NUM + i;               // + k*XK
  const float* vp = v + j * (BATCH * OUT_DIM) + mn;   // + k*OUT_DIM

  v8f acc = {};
  for (int k0 = 0; k0 < BATCH; k0 += 4) {
    int k = k0 + 2 * hi;
    v2f a, b;
    a.x = xp[k * XK];
    a.y = xp[(k + 1) * XK];
    b.x = vp[k * OUT_DIM];
    b.y = vp[(k + 1) * OUT_DIM];
    acc = wmma_f32_k4(a, b, acc);
  }
  for (int r = 0; r < 8; ++r) {
    int row = r0 + r + 8 * hi;
    P[row * PN + j * OUT_DIM + mn] = acc[r];
  }
}

// b[i,j] += (1/BATCH) * sum_{d,o} W[i,j,o,d] * P[(i*8+d), (j*16+o)]
__global__ void k_agree(const float* __restrict__ W, const float* __restrict__ P,
                        float* __restrict__ bij) {
  int idx = blockIdx.x * blockDim.x + threadIdx.x;
  if (idx >= IN_NUM * OUT_NUM) return;
  int i = idx / OUT_NUM, j = idx % OUT_NUM;
  const float* wp = W + (i * OUT_NUM + j) * OUT_DIM * IN_DIM;  // [o][d]
  const float* pp = P + (i * IN_DIM) * PN + j * OUT_DIM;       // [d][o], row stride PN
  float sum = 0.0f;
  for (int o = 0; o < OUT_DIM; ++o)
    for (int d = 0; d < IN_DIM; ++d)
      sum += wp[o * IN_DIM + d] * pp[d * PN + o];
  bij[idx] += sum * (1.0f / BATCH);
}

extern "C" void kernel_launch(void* const* d_in, const int* in_sizes, int n_in,
                              void* d_out, int out_size, void* d_ws, size_t ws_size,
                              hipStream_t stream) {
  const float* x = (const float*)d_in[0];  // [256, 8, 1152]
  const float* W = (const float*)d_in[1];  // [1152, 10, 16, 8]
  float* v  = (float*)d_out;               // [10, 256, 16]
  float* ws = (float*)d_ws;
  float* c   = ws;                          // 11520 floats
  float* bij = ws + 11520;                  // 11520 floats
  float* sp  = ws + 23040;                  // NSPLIT * 40960 = 368640 floats
  float* s   = ws + 23040 + NSPLIT * S_ELEMS;          // 40960 floats
  float* P   = ws + 23040 + NSPLIT * S_ELEMS + S_ELEMS; // 1474560 floats

  k_zero<<<(IN_NUM * OUT_NUM + 255) / 256, 256, 0, stream>>>(bij, IN_NUM * OUT_NUM);
  for (int it = 0; it < 3; ++it) {
    k_softmax<<<(IN_NUM + 255) / 256, 256, 0, stream>>>(bij, c);
    // grid: 4 batch-tile groups (x4 waves) x 10 capsules x 9 i-splits of 128
    k_s_gemm<<<dim3(4, 10, NSPLIT), 128, 0, stream>>>(x, W, c, sp);
    k_reduce_s<<<(S_ELEMS + 255) / 256, 256, 0, stream>>>(sp, s);
    k_squash<<<(OUT_NUM * BATCH + 255) / 256, 256, 0, stream>>>(s, v);
    if (it < 2) {
      // grid: 9216/16/4 = 144 row-tile groups x 10 capsules
      k_p_gemm<<<dim3(144, 10), 128, 0, stream>>>(x, v, P);
      k_agree<<<(IN_NUM * OUT_NUM + 255) / 256, 256, 0, stream>>>(W, P, bij);
    }
  }
}